// ImprovedSpectralFilter_10642928959587
// MI455X (gfx1250) — compile-verified
//
#include <hip/hip_runtime.h>
#include <hip/hip_bf16.h>
#include <math.h>
#include <stdint.h>

// Problem constants (match reference)
#define B_    64
#define NPG_  2048
#define KMAX_ 64
#define DIN_  256
#define DOUT_ 256
#define EDIM_ 16

typedef __attribute__((ext_vector_type(2))) float v2f;
typedef __attribute__((ext_vector_type(8))) float v8f;
typedef unsigned int u32x4 __attribute__((ext_vector_type(4)));
typedef int i32x4 __attribute__((ext_vector_type(4)));
typedef int i32x8 __attribute__((ext_vector_type(8)));

// D = A(16x4 f32) * B(4x16 f32) + C(16x16 f32)
static __device__ __forceinline__ v8f wmma4(v2f a, v2f b, v8f c) {
  return __builtin_amdgcn_wmma_f32_16x16x4_f32(
      /*neg_a=*/false, a, /*neg_b=*/false, b,
      /*c_mod=*/(short)0, c, /*reuse_a=*/false, /*reuse_b=*/false);
}

// ---------------------------------------------------------------------------
// TDM: 2D tile load Global -> LDS with row padding (D# per CDNA5 ISA
// §8). data_size = 4 bytes. pad codes: interval v -> 2^(v+1) DWORDs between
// pads, amount v -> (v+1) DWORDs of pad. Issued once per executing wave;
// tracked with TENSORcnt.
// ---------------------------------------------------------------------------
static __device__ __forceinline__ void tdm_load_2d(
    unsigned lds_off, const void* gptr,
    unsigned tensor_d0, unsigned tensor_d1, unsigned stride0,
    unsigned tile_d0, unsigned tile_d1,
    unsigned pad_interval_code, unsigned pad_amount_code) {
  const unsigned long long ga = (unsigned long long)(uintptr_t)gptr;
  u32x4 g0;
  g0.x = 1u;                                        // count=1, user D#
  g0.y = lds_off;                                   // lds_addr (bytes)
  g0.z = (unsigned)(ga & 0xFFFFFFFFu);              // global_addr[31:0]
  g0.w = (unsigned)((ga >> 32) & 0x01FFFFFFu) | (2u << 30);  // addr[56:32], type=2
  i32x8 g1;
  g1[0] = (int)((2u << 16) | (1u << 20) |           // data_size=4B, pad_enable
                (pad_interval_code << 22) | (pad_amount_code << 25));
  g1[1] = (int)((tensor_d0 & 0xFFFFu) << 16);       // tensor_dim0[15:0]
  g1[2] = (int)((tensor_d0 >> 16) | ((tensor_d1 & 0xFFFFu) << 16));
  g1[3] = (int)((tensor_d1 >> 16) | (tile_d0 << 16));
  g1[4] = (int)(tile_d1 & 0xFFFFu);                 // tile_dim1, tile_dim2=0
  g1[5] = (int)stride0;                             // tensor_dim0_stride[31:0]
  g1[6] = 0;
  g1[7] = 0;
  const i32x4 z4 = {0, 0, 0, 0};
#if defined(__clang_major__) && (__clang_major__ >= 23)
  const i32x8 z8 = {0, 0, 0, 0, 0, 0, 0, 0};
  __builtin_amdgcn_tensor_load_to_lds(g0, g1, z4, z4, z8, 0);
#else
  __builtin_amdgcn_tensor_load_to_lds(g0, g1, z4, z4, 0);
#endif
}

extern __shared__ float lds[];

// ---------------------------------------------------------------------------
// Kernel 1: eig_encoder + 2-head self-attention + filter MLP.
// One block per batch, 64 threads (1 token per thread, 2 waves).
// ---------------------------------------------------------------------------
__global__ __launch_bounds__(64) void filter_kernel(
    const float* __restrict__ eigenvalues, const unsigned char* __restrict__ eig_mask,
    const float* __restrict__ W1, const float* __restrict__ b1,
    const float* __restrict__ g1, const float* __restrict__ be1,
    const float* __restrict__ W2, const float* __restrict__ b2,
    const float* __restrict__ Wq, const float* __restrict__ bq,
    const float* __restrict__ Wk, const float* __restrict__ bk,
    const float* __restrict__ Wv, const float* __restrict__ bv,
    const float* __restrict__ Wo, const float* __restrict__ bo,
    const float* __restrict__ Wf1, const float* __restrict__ bf1,
    const float* __restrict__ Wf2, const float* __restrict__ bf2,
    float* __restrict__ filt) {
  __shared__ float kbuf[KMAX_][EDIM_];
  __shared__ float vbuf[KMAX_][EDIM_];
  __shared__ float msk[KMAX_];

  const int b = blockIdx.x;
  const int t = threadIdx.x;

  const float ev = eigenvalues[b * KMAX_ + t];
  const float mk = eig_mask[b * KMAX_ + t] ? 1.f : 0.f;
  msk[t] = mk;

  // eig_encoder: Linear(1,16) -> LN -> ReLU -> Linear(16,16)
  float e[EDIM_];
  float mean = 0.f;
  #pragma unroll
  for (int j = 0; j < EDIM_; ++j) { e[j] = ev * W1[j] + b1[j]; mean += e[j]; }
  mean *= (1.f / EDIM_);
  float var = 0.f;
  #pragma unroll
  for (int j = 0; j < EDIM_; ++j) { float d = e[j] - mean; var += d * d; }
  var *= (1.f / EDIM_);
  const float inv = rsqrtf(var + 1e-5f);
  float h1[EDIM_];
  #pragma unroll
  for (int j = 0; j < EDIM_; ++j)
    h1[j] = fmaxf((e[j] - mean) * inv * g1[j] + be1[j], 0.f);
  float h[EDIM_];
  #pragma unroll
  for (int j = 0; j < EDIM_; ++j) {
    float a = b2[j];
    #pragma unroll
    for (int i = 0; i < EDIM_; ++i) a += h1[i] * W2[i * EDIM_ + j];
    h[j] = a;
  }

  // q/k/v projections
  float q[EDIM_];
  #pragma unroll
  for (int j = 0; j < EDIM_; ++j) {
    float aq = bq[j], ak = bk[j], av = bv[j];
    #pragma unroll
    for (int i = 0; i < EDIM_; ++i) {
      aq += h[i] * Wq[i * EDIM_ + j];
      ak += h[i] * Wk[i * EDIM_ + j];
      av += h[i] * Wv[i * EDIM_ + j];
    }
    q[j] = aq;
    kbuf[t][j] = ak;
    vbuf[t][j] = av;
  }
  __syncthreads();

  // attention (two passes: max, then exp/accumulate)
  float ctx[EDIM_];
  const float scale = 0.3535533905932738f;  // 1/sqrt(8)
  #pragma unroll
  for (int hh = 0; hh < 2; ++hh) {
    const int base = hh * 8;
    float mx = -1e30f;
    for (int s = 0; s < KMAX_; ++s) {
      float d = 0.f;
      #pragma unroll
      for (int dd = 0; dd < 8; ++dd) d += q[base + dd] * kbuf[s][base + dd];
      d = (msk[s] != 0.f) ? d * scale : -1e9f;
      mx = fmaxf(mx, d);
    }
    float denom = 0.f;
    float acc[8] = {0.f, 0.f, 0.f, 0.f, 0.f, 0.f, 0.f, 0.f};
    for (int s = 0; s < KMAX_; ++s) {
      float d = 0.f;
      #pragma unroll
      for (int dd = 0; dd < 8; ++dd) d += q[base + dd] * kbuf[s][base + dd];
      d = (msk[s] != 0.f) ? d * scale : -1e9f;
      const float w = __expf(d - mx);
      denom += w;
      #pragma unroll
      for (int dd = 0; dd < 8; ++dd) acc[dd] += w * vbuf[s][base + dd];
    }
    const float rden = 1.f / denom;
    #pragma unroll
    for (int dd = 0; dd < 8; ++dd) ctx[base + dd] = acc[dd] * rden;
  }

  // out proj + filter MLP: Linear(16,32) ReLU Linear(32,1) Tanh
  float c2[EDIM_];
  #pragma unroll
  for (int j = 0; j < EDIM_; ++j) {
    float a = bo[j];
    #pragma unroll
    for (int i = 0; i < EDIM_; ++i) a += ctx[i] * Wo[i * EDIM_ + j];
    c2[j] = a;
  }
  float fo = bf2[0];
  #pragma unroll
  for (int j = 0; j < 32; ++j) {
    float a = bf1[j];
    #pragma unroll
    for (int i = 0; i < EDIM_; ++i) a += c2[i] * Wf1[i * 32 + j];
    fo += fmaxf(a, 0.f) * Wf2[j];
  }
  filt[b * KMAX_ + t] = tanhf(fo) * mk;
}

// ---------------------------------------------------------------------------
// Kernel 2: F[b] = diag(filt[b]) * (V_b^T * X_b)   [64 x 256] per batch.
// Grid (B, 4). TDM double-buffers 64x64 tiles of V^T and X into LDS
// (row stride 72 floats via TDM padding -> conflict-free ds reads).
// Wave 0 issues the DMAs; chunk c+1 DMA overlaps chunk c WMMA.
// ---------------------------------------------------------------------------
#define K2ST 72                  // LDS row stride (floats) in kernel 2 tiles
#define K2TS (64 * K2ST)         // one tile (floats)

__global__ __launch_bounds__(128) void xfreq_kernel(
    const float* __restrict__ x, const float* __restrict__ V,
    const float* __restrict__ filt, float* __restrict__ F) {
  const int b = blockIdx.x;
  const int nb = blockIdx.y * 64;
  const int w = threadIdx.x >> 5;
  const int lane = threadIdx.x & 31;
  const int half = lane >> 4;
  const int l16 = lane & 15;
  const int m0 = w * 16;

  const float* __restrict__ Vb = V + (size_t)b * NPG_ * KMAX_;
  const float* __restrict__ Xb = x + (size_t)b * NPG_ * DIN_;

  // buffers: p in {0,1}: [VT tile][XT tile]
  if (w == 0) {
    tdm_load_2d((unsigned)(uintptr_t)(lds), Vb,
                KMAX_, NPG_, KMAX_, 64, 64, /*pad*/ 5, 7);
    tdm_load_2d((unsigned)(uintptr_t)(lds + K2TS), Xb + nb,
                DIN_, NPG_, DIN_, 64, 64, /*pad*/ 5, 7);
  }

  v8f acc[4] = {};
  const int C = NPG_ / 64;  // 32 K-chunks
  for (int c = 0; c < C; ++c) {
    const int p = c & 1;
    if (w == 0) {
      if (c + 1 < C) {
        const int q = p ^ 1;
        tdm_load_2d((unsigned)(uintptr_t)(lds + q * 2 * K2TS),
                    Vb + (size_t)(c + 1) * 64 * KMAX_,
                    KMAX_, NPG_, KMAX_, 64, 64, 5, 7);
        tdm_load_2d((unsigned)(uintptr_t)(lds + q * 2 * K2TS + K2TS),
                    Xb + (size_t)(c + 1) * 64 * DIN_ + nb,
                    DIN_, NPG_, DIN_, 64, 64, 5, 7);
        __builtin_amdgcn_s_wait_tensorcnt(2);  // chunk c's 2 loads done
      } else {
        __builtin_amdgcn_s_wait_tensorcnt(0);
      }
    }
    __syncthreads();

    const float* VT = lds + p * 2 * K2TS;
    const float* XT = VT + K2TS;
    for (int kk = 0; kk < 64; kk += 4) {
      const int ka = kk + 2 * half;
      v2f a;
      a.x = VT[ka * K2ST + m0 + l16];          // A = V^T tile (k-major rows)
      a.y = VT[(ka + 1) * K2ST + m0 + l16];
      #pragma unroll
      for (int t = 0; t < 4; ++t) {
        v2f bb;
        bb.x = XT[ka * K2ST + t * 16 + l16];
        bb.y = XT[(ka + 1) * K2ST + t * 16 + l16];
        acc[t] = wmma4(a, bb, acc[t]);
      }
    }
    __syncthreads();  // all waves done reading before buffer p is re-filled
  }

  // epilogue: scale row m by filt[b][m], store F[b] row-major [64,256]
  #pragma unroll
  for (int r = 0; r < 8; ++r) {
    const int row = m0 + r + 8 * half;
    const float fs = filt[b * KMAX_ + row];
    #pragma unroll
    for (int t = 0; t < 4; ++t) {
      F[((size_t)b * KMAX_ + row) * DIN_ + nb + t * 16 + l16] = acc[t][r] * fs;
    }
  }
}

// ---------------------------------------------------------------------------
// Kernel 3 (fused): Y = V_b * F[b]; OUT = LayerNorm(Y @ Wp + bp).
// Grid (B, 32). TDM stages the F tile (stride 268) and the 64-node V chunk
// (stride 68) into LDS; strides chosen for conflict-free WMMA fragment reads.
// ---------------------------------------------------------------------------
#define FST 268  // Fsh/Ysh LDS row stride (floats): 256 + 12 pad
#define VST 68   // Vsh LDS row stride (floats): 64 + 4 pad

__global__ __launch_bounds__(128) void spectral_kernel(
    const float* __restrict__ V, const float* __restrict__ F,
    const float* __restrict__ Wp, const float* __restrict__ bp,
    const float* __restrict__ gp, const float* __restrict__ bep,
    float* __restrict__ out) {
  float* Fsh = lds;                          // 64 * 268
  float* Ysh = Fsh + KMAX_ * FST;            // 64 * 268
  float* Vsh = Ysh + 64 * FST;               // 64 * 68

  const int b = blockIdx.x;
  const int mg = blockIdx.y;
  const int tid = threadIdx.x;
  const int w = tid >> 5;
  const int lane = tid & 31;
  const int half = lane >> 4;
  const int l16 = lane & 15;

  const float* __restrict__ Fb = F + (size_t)b * KMAX_ * DIN_;
  const float* __restrict__ Vb = V + (size_t)b * NPG_ * KMAX_;

  if (w == 0) {
    // F[b]: 64 rows x 256 cols, pad 12 DW per 256 DW -> stride 268
    tdm_load_2d((unsigned)(uintptr_t)Fsh, Fb,
                DIN_, KMAX_, DIN_, 256, 64, /*pad*/ 7, 11);
    // V chunk: 64 node rows x 64 eig cols, pad 4 DW per 64 DW -> stride 68
    tdm_load_2d((unsigned)(uintptr_t)Vsh, Vb + (size_t)mg * 64 * KMAX_,
                KMAX_, NPG_, KMAX_, 64, 64, /*pad*/ 5, 3);
    __builtin_amdgcn_s_wait_tensorcnt(0);
  }
  __syncthreads();

  // -------- phase 1: Y strip (16x256) = Vstrip (16x64) * F (64x256) --------
  v8f acc[16] = {};
  for (int kk = 0; kk < KMAX_; kk += 4) {
    const int ka = kk + 2 * half;
    v2f a;
    a.x = Vsh[(w * 16 + l16) * VST + ka];
    a.y = Vsh[(w * 16 + l16) * VST + ka + 1];
    #pragma unroll
    for (int nt = 0; nt < 16; ++nt) {
      v2f bb;
      bb.x = Fsh[ka * FST + nt * 16 + l16];
      bb.y = Fsh[(ka + 1) * FST + nt * 16 + l16];
      acc[nt] = wmma4(a, bb, acc[nt]);
    }
  }
  // stash Y strip in LDS
  #pragma unroll
  for (int nt = 0; nt < 16; ++nt) {
    #pragma unroll
    for (int r = 0; r < 8; ++r) {
      Ysh[(w * 16 + r + 8 * half) * FST + nt * 16 + l16] = acc[nt][r];
    }
  }
  __syncthreads();

  // -------- phase 2: OUT strip (16x256) = Ystrip (16x256) * Wp (256x256) ---
  v8f acc2[16] = {};
  const int ybase = (w * 16 + l16) * FST;
  for (int kk = 0; kk < DIN_; kk += 4) {
    const int ka = kk + 2 * half;
    v2f a;
    a.x = Ysh[ybase + ka];
    a.y = Ysh[ybase + ka + 1];
    #pragma unroll
    for (int nt = 0; nt < 16; ++nt) {
      v2f bb;
      bb.x = Wp[(size_t)ka * DOUT_ + nt * 16 + l16];
      bb.y = Wp[(size_t)(ka + 1) * DOUT_ + nt * 16 + l16];
      acc2[nt] = wmma4(a, bb, acc2[nt]);
    }
  }

  // bias add, park rows in (now-free) Fsh region for row-wise LayerNorm
  float* Osh = Fsh + (w * 16) * FST;
  #pragma unroll
  for (int nt = 0; nt < 16; ++nt) {
    const float bias = bp[nt * 16 + l16];
    #pragma unroll
    for (int r = 0; r < 8; ++r) {
      Osh[(r + 8 * half) * FST + nt * 16 + l16] = acc2[nt][r] + bias;
    }
  }
  __syncthreads();

  // LayerNorm per row (full wave reduces one row at a time), coalesced store
  #pragma unroll 1
  for (int r = 0; r < 16; ++r) {
    const float* rowp = Osh + r * FST;
    float s = 0.f, s2 = 0.f;
    #pragma unroll
    for (int i = 0; i < 8; ++i) {
      const float v = rowp[lane + 32 * i];
      s += v;
      s2 += v * v;
    }
    #pragma unroll
    for (int off = 16; off > 0; off >>= 1) {
      s += __shfl_xor(s, off, 32);
      s2 += __shfl_xor(s2, off, 32);
    }
    const float mean = s * (1.f / DOUT_);
    const float var = s2 * (1.f / DOUT_) - mean * mean;
    const float inv = rsqrtf(var + 1e-5f);
    const size_t grow = ((size_t)b * NPG_ + mg * 64 + w * 16 + r) * (size_t)DOUT_;
    #pragma unroll
    for (int i = 0; i < 8; ++i) {
      const int c = lane + 32 * i;
      out[grow + c] = (rowp[c] - mean) * inv * gp[c] + bep[c];
    }
  }
}

// ---------------------------------------------------------------------------
extern "C" void kernel_launch(void* const* d_in, const int* in_sizes, int n_in,
                              void* d_out, int out_size, void* d_ws, size_t ws_size,
                              hipStream_t stream) {
  (void)in_sizes; (void)n_in; (void)out_size; (void)ws_size;

  const float* x   = (const float*)d_in[0];
  const float* V   = (const float*)d_in[1];
  const float* ev  = (const float*)d_in[2];
  const unsigned char* mask = (const unsigned char*)d_in[3];
  // d_in[4] = batch (unused, equal-size graphs), d_in[5] = bsz (compile-time)
  const float* W1 = (const float*)d_in[6];
  const float* b1 = (const float*)d_in[7];
  const float* g1 = (const float*)d_in[8];
  const float* be1= (const float*)d_in[9];
  const float* W2 = (const float*)d_in[10];
  const float* b2 = (const float*)d_in[11];
  const float* Wq = (const float*)d_in[12];
  const float* bq = (const float*)d_in[13];
  const float* Wk = (const float*)d_in[14];
  const float* bk = (const float*)d_in[15];
  const float* Wv = (const float*)d_in[16];
  const float* bv = (const float*)d_in[17];
  const float* Wo = (const float*)d_in[18];
  const float* bo = (const float*)d_in[19];
  const float* Wf1= (const float*)d_in[20];
  const float* bf1= (const float*)d_in[21];
  const float* Wf2= (const float*)d_in[22];
  const float* bf2= (const float*)d_in[23];
  const float* Wp = (const float*)d_in[24];
  const float* bp = (const float*)d_in[25];
  const float* gp = (const float*)d_in[26];
  const float* bep= (const float*)d_in[27];

  float* out  = (float*)d_out;
  float* filt = (float*)d_ws;                        // B*64 floats
  float* F    = (float*)((char*)d_ws + 32768);       // B*64*256 floats (4 MB)

  // 1) per-eigenvalue filter coefficients
  filter_kernel<<<dim3(B_), dim3(KMAX_), 0, stream>>>(
      ev, mask, W1, b1, g1, be1, W2, b2,
      Wq, bq, Wk, bk, Wv, bv, Wo, bo, Wf1, bf1, Wf2, bf2, filt);

  // 2) scaled spectral coefficients F = diag(filt) V^T X  (TDM double-buffer)
  const size_t lds2 = (size_t)(4 * K2TS) * sizeof(float);  // 2 bufs x (VT+XT)
  xfreq_kernel<<<dim3(B_, DIN_ / 64), dim3(128), lds2, stream>>>(x, V, filt, F);

  // 3) fused synthesis + projection + LayerNorm (TDM-staged F and V tiles)
  const size_t lds3 = (size_t)(KMAX_ * FST + 64 * FST + 64 * VST) * sizeof(float);
  spectral_kernel<<<dim3(B_, NPG_ / 64), dim3(128), lds3, stream>>>(
      V, F, Wp, bp, gp, bep, out);
}